// GATReduce_1451698946380
// MI455X (gfx1250) — compile-verified
//
#include <hip/hip_runtime.h>

// GAT-style attention reduce for MI455X (gfx1250, wave32).
// One wave per node; all global traffic is coalesced b128; the per-deg
// sum-over-D reduction runs through V_WMMA_F32_16X16X4_F32 with an all-ones
// B matrix (WMMA as a 32-lane reduction engine, full fp32 precision).

typedef float v2f __attribute__((ext_vector_type(2)));
typedef float v8f __attribute__((ext_vector_type(8)));

#define DEG 32
#define DIM 128
#define WPB 8      // waves per block (256 threads)
#define PS_LD 33   // padded LDS row stride (bank-conflict avoidance)

__global__ __launch_bounds__(256) void gat_reduce_kernel(
    const float* __restrict__ a1, const float* __restrict__ a2,
    const float* __restrict__ ft, float* __restrict__ out, int n_nodes)
{
    __shared__ float ps_all[WPB][DEG * PS_LD]; // per-(deg,lane) partial sums
    __shared__ float av_all[WPB][DEG];         // reduced a[deg]

    const int lane = threadIdx.x & 31;
    const int wave = threadIdx.x >> 5;
    const int node = blockIdx.x * WPB + wave;
    if (node >= n_nodes) return; // wave-uniform guard (EXEC stays all-1s)

    float* ps = ps_all[wave];
    float* av = av_all[wave];

    const size_t row = (size_t)node * DEG * DIM;
    const float* __restrict__ ftn = ft + row;

    // Prefetch the ft stream (16KB/node) while we process a1/a2.
    #pragma unroll
    for (int i = 0; i < 4; ++i)
        __builtin_prefetch((const char*)ftn + ((i << 5) + lane) * 128, 0, 1);

    // ---- phase 0: s1 = rowsum(a1[node])  (coalesced b128 + wave32 butterfly)
    const float4* a1v = (const float4*)(a1 + (size_t)node * DIM);
    float4 v0 = a1v[lane];
    float s1 = v0.x + v0.y + v0.z + v0.w;
    #pragma unroll
    for (int off = 16; off >= 1; off >>= 1)
        s1 += __shfl_xor(s1, off, 32);

    // ---- phase 1: per-(deg,lane) partial sums of a2 into LDS
    const float4* a2v = (const float4*)(a2 + row);
    #pragma unroll 4
    for (int deg = 0; deg < DEG; ++deg) {
        float4 w = a2v[deg * (DIM / 4) + lane];
        ps[deg * PS_LD + lane] = w.x + w.y + w.z + w.w;
    }

    // ---- WMMA reduction: A[m=deg][k=lane] x ones(4x16) accumulated over 8
    // K-steps -> every C column holds rowsum over the 32 lane-partials.
    // f32 16x16x4 A layout: lanes 0-15 hold K=0,1; lanes 16-31 hold K=2,3.
    const int m    = lane & 15;
    const int koff = (lane >> 4) * 2;
    v2f bones; bones[0] = 1.0f; bones[1] = 1.0f;
    v8f c0 = {}; v8f c1 = {};
    #pragma unroll
    for (int kk = 0; kk < 8; ++kk) {
        const int kb = kk * 4 + koff;
        v2f ta, tb;
        ta[0] = ps[m * PS_LD + kb];
        ta[1] = ps[m * PS_LD + kb + 1];
        tb[0] = ps[(16 + m) * PS_LD + kb];
        tb[1] = ps[(16 + m) * PS_LD + kb + 1];
        c0 = __builtin_amdgcn_wmma_f32_16x16x4_f32(false, ta, false, bones,
                                                   (short)0, c0, false, false);
        c1 = __builtin_amdgcn_wmma_f32_16x16x4_f32(false, tb, false, bones,
                                                   (short)0, c1, false, false);
    }

    // C layout: lanes 0-15 hold M=r in VGPR r, lanes 16-31 hold M=8+r.
    // Columns are identical (B all-ones), so lanes with N==0 scatter to LDS.
    if ((lane & 15) == 0) {
        const int mb = (lane >> 4) * 8;
        #pragma unroll
        for (int r = 0; r < 8; ++r) {
            av[mb + r]      = c0[r];
            av[16 + mb + r] = c1[r];
        }
    }

    // ---- softmax over DEG (lane == deg), leaky_relu slope 0.01
    float aval = av[lane] + s1;
    float al = aval > 0.0f ? aval : 0.01f * aval;
    float mx = al;
    #pragma unroll
    for (int off = 16; off >= 1; off >>= 1)
        mx = fmaxf(mx, __shfl_xor(mx, off, 32));
    float ex = __expf(al - mx);
    float sm = ex;
    #pragma unroll
    for (int off = 16; off >= 1; off >>= 1)
        sm += __shfl_xor(sm, off, 32);
    float e = ex / sm;

    // ---- phase 2: out[node,:] = sum_deg e[deg] * ft[node,deg,:]
    const float4* ftv = (const float4*)ftn;
    float4 acc = make_float4(0.f, 0.f, 0.f, 0.f);
    #pragma unroll 4
    for (int deg = 0; deg < DEG; ++deg) {
        float ed = __shfl(e, deg, 32); // broadcast e[deg] from lane==deg
        float4 f = ftv[deg * (DIM / 4) + lane];
        acc.x += ed * f.x; acc.y += ed * f.y;
        acc.z += ed * f.z; acc.w += ed * f.w;
    }
    ((float4*)(out + (size_t)node * DIM))[lane] = acc;
}

extern "C" void kernel_launch(void* const* d_in, const int* in_sizes, int n_in,
                              void* d_out, int out_size, void* d_ws, size_t ws_size,
                              hipStream_t stream) {
    const float* a1 = (const float*)d_in[0];
    const float* a2 = (const float*)d_in[1];
    const float* ft = (const float*)d_in[2];
    float* out = (float*)d_out;

    const int n_nodes = in_sizes[0] / DIM;           // a1 is [N, 128]
    const int blocks  = (n_nodes + WPB - 1) / WPB;   // 8 waves (nodes) per block
    gat_reduce_kernel<<<blocks, 256, 0, stream>>>(a1, a2, ft, out, n_nodes);
}